// GASLayer_63754494542503
// MI455X (gfx1250) — compile-verified
//
#include <hip/hip_runtime.h>
#include <math.h>

// Problem shape (fixed by reference)
#define T_DIM   16384
#define D_DIM   2048
#define NCHUNK  128
#define CHUNK   128            // T_DIM / NCHUNK

#define ETA_MU_F  0.05f
#define ETA_VAR_F 0.05f
#define DECAY_F   0.95f        // 1 - eta (same for mu and var)
#define EPS_F     1e-13f

typedef float v2f __attribute__((ext_vector_type(2)));
typedef float v8f __attribute__((ext_vector_type(8)));

// ---------------------------------------------------------------------------
// Pass 1: per-chunk sum, sum^2, and mu-aggregate M_c = sum_k a^{n-1-k} eta x_k
// (computed as the recurrence with zero init). Streaming, fully coalesced.
// ---------------------------------------------------------------------------
__global__ void k_chunk_stats(const float* __restrict__ x,
                              float* __restrict__ ws_sum,
                              float* __restrict__ ws_sum2,
                              float* __restrict__ ws_M) {
    const int col = blockIdx.x * blockDim.x + threadIdx.x;
    const int c   = blockIdx.y;
    const float* xp = x + (size_t)c * CHUNK * D_DIM + col;
    float s = 0.f, s2 = 0.f, m = 0.f;
#pragma unroll 4
    for (int k = 0; k < CHUNK; ++k) {
        float v = xp[(size_t)k * D_DIM];
        __builtin_prefetch(&xp[(size_t)(k + 16) * D_DIM], 0, 1); // global_prefetch
        s  += v;
        s2  = fmaf(v, v, s2);
        m   = fmaf(ETA_MU_F, v - m, m);     // m += eta*(v - m)
    }
    ws_sum [c * D_DIM + col] = s;
    ws_sum2[c * D_DIM + col] = s2;
    ws_M   [c * D_DIM + col] = m;
}

// ---------------------------------------------------------------------------
// Combine kernel (WMMA block-scan). For each column d:
//   entry[0]   = init (mu0 or var0, from ws_sum/ws_sum2)
//   entry[c+1] = A*entry[c] + agg[c],  A = 0.95^CHUNK
// Done 16 chunks at a time: P = L*G with L strictly lower triangular,
// L(i,j) = A^{i-1-j} (j<i), via 4x v_wmma_f32_16x16x4_f32; then
// entry[16g+i] = P[i] + A^i * carry;  carry' = A^16*carry + A*P[15] + agg[15].
// One wave handles a 16-column tile; 8 waves per block.
// ---------------------------------------------------------------------------
__global__ void k_combine_wmma(const float* __restrict__ ws_sum,
                               const float* __restrict__ ws_sum2,
                               const float* __restrict__ agg,    // M or V
                               float* __restrict__ entry,        // muin or varin
                               int is_var,
                               float A, float A8, float A16, float log2A) {
    const int wave = threadIdx.x >> 5;
    const int lane = threadIdx.x & 31;
    const int n    = lane & 15;           // column within tile / A-operand row M
    const int h    = lane >> 4;           // half-wave selector
    const int tile = blockIdx.x * 8 + wave;
    const int col  = tile * 16 + n;

    // init value for this column (mu0 / var0 from global sums)
    float s = 0.f, s2 = 0.f;
    for (int c = 0; c < NCHUNK; ++c) {
        s  += ws_sum [c * D_DIM + col];
        s2 += ws_sum2[c * D_DIM + col];
    }
    const float Tf  = (float)T_DIM;
    float mu0  = s / Tf;
    float var0 = fmaf(-mu0, s, s2) / (Tf - 1.0f);   // unbiased var
    float carry = is_var ? var0 : mu0;

    // L-matrix lane constants (A-operand 16x4 layout: lane holds row M=n,
    // VGPR v holds K = 4*kt + v + 2*h).  L(m,j) = (j<m) ? A^{m-1-j} : 0.
    v2f Lk[4];
#pragma unroll
    for (int kt = 0; kt < 4; ++kt) {
        int j0 = 4 * kt + 2 * h;
        int j1 = j0 + 1;
        Lk[kt].x = (j0 < n) ? exp2f((float)(n - 1 - j0) * log2A) : 0.f;
        Lk[kt].y = (j1 < n) ? exp2f((float)(n - 1 - j1) * log2A) : 0.f;
    }

    for (int g = 0; g < NCHUNK / 16; ++g) {
        const float* gb = agg + (size_t)(16 * g) * D_DIM + col;
        v8f acc = {0.f, 0.f, 0.f, 0.f, 0.f, 0.f, 0.f, 0.f};
#pragma unroll
        for (int kt = 0; kt < 4; ++kt) {
            // B-operand 4x16: VGPR v holds K = 4*kt + v + 2*h, N = n
            v2f b;
            b.x = gb[(size_t)(4 * kt + 2 * h)     * D_DIM];
            b.y = gb[(size_t)(4 * kt + 2 * h + 1) * D_DIM];
            acc = __builtin_amdgcn_wmma_f32_16x16x4_f32(
                false, Lk[kt], false, b, (short)0, acc, false, false);
        }
        // C/D layout: VGPR r holds row M = r + 8*h, column N = n
        float pm = h ? A8 : 1.0f;            // A^(8h), times A^r below
#pragma unroll
        for (int r = 0; r < 8; ++r) {
            int row = 16 * g + r + 8 * h;
            entry[(size_t)row * D_DIM + col] = fmaf(pm, carry, acc[r]);
            pm *= A;
        }
        // carry' = A^16*carry + A*P[15] + agg[16g+15]; P[15] is acc[7] on h==1
        float last = gb[(size_t)15 * D_DIM];
        float cn   = fmaf(A16, carry, fmaf(A, acc[7], last));
        carry = __shfl(cn, 16 + n, 32);      // broadcast from upper half, same N
    }
}

// ---------------------------------------------------------------------------
// Pass 3: replay mu inside each chunk, accumulate var-aggregate
// V_c = sum_k a^{n-1-k} eta_v (x_k - mu_k)^2  (recurrence, zero init).
// ---------------------------------------------------------------------------
__global__ void k_var_agg(const float* __restrict__ x,
                          const float* __restrict__ ws_muin,
                          float* __restrict__ ws_V) {
    const int col = blockIdx.x * blockDim.x + threadIdx.x;
    const int c   = blockIdx.y;
    const float* xp = x + (size_t)c * CHUNK * D_DIM + col;
    float mu = ws_muin[c * D_DIM + col];
    float V  = 0.f;
#pragma unroll 4
    for (int k = 0; k < CHUNK; ++k) {
        float v = xp[(size_t)k * D_DIM];
        mu = fmaf(ETA_MU_F, v - mu, mu);
        float d = v - mu;
        V = fmaf(DECAY_F, V, ETA_VAR_F * (d * d));
    }
    ws_V[c * D_DIM + col] = V;
}

// ---------------------------------------------------------------------------
// Pass 5: replay mu+var per chunk (exact reference update formulas), emit
// norm / mu / var with non-temporal stores (keep x resident in L2).
// ---------------------------------------------------------------------------
__global__ void k_final(const float* __restrict__ x,
                        const float* __restrict__ ws_muin,
                        const float* __restrict__ ws_varin,
                        float* __restrict__ out_norm,
                        float* __restrict__ out_info) {
    const int col = blockIdx.x * blockDim.x + threadIdx.x;
    const int c   = blockIdx.y;
    const size_t trow = (size_t)c * CHUNK;
    const float* xp = x + trow * D_DIM + col;
    float mu  = ws_muin [c * D_DIM + col];
    float var = ws_varin[c * D_DIM + col];
#pragma unroll 2
    for (int k = 0; k < CHUNK; ++k) {
        float v = xp[(size_t)k * D_DIM];
        mu = fmaf(ETA_MU_F, v - mu, mu);                  // mu += eta*(x - mu)
        float d = v - mu;
        var = fmaf(DECAY_F, var, ETA_VAR_F * (d * d));    // var = .95*var + .05*d^2
        float nrm = d / (sqrtf(var) + EPS_F);
        size_t t = trow + (size_t)k;
        __builtin_nontemporal_store(nrm, out_norm + t * D_DIM + col);
        __builtin_nontemporal_store(mu,  out_info + t * (2 * D_DIM) + col);
        __builtin_nontemporal_store(var, out_info + t * (2 * D_DIM) + D_DIM + col);
    }
}

// ---------------------------------------------------------------------------
extern "C" void kernel_launch(void* const* d_in, const int* in_sizes, int n_in,
                              void* d_out, int out_size, void* d_ws, size_t ws_size,
                              hipStream_t stream) {
    (void)in_sizes; (void)n_in; (void)out_size; (void)ws_size;
    const float* x = (const float*)d_in[0];
    float* out      = (float*)d_out;
    float* out_norm = out;
    float* out_info = out + (size_t)T_DIM * D_DIM;

    float* ws = (float*)d_ws;
    const size_t SL = (size_t)NCHUNK * D_DIM;       // 256 KiB each (floats)
    float* ws_sum   = ws + 0 * SL;
    float* ws_sum2  = ws + 1 * SL;
    float* ws_M     = ws + 2 * SL;
    float* ws_V     = ws + 3 * SL;
    float* ws_muin  = ws + 4 * SL;
    float* ws_varin = ws + 5 * SL;                   // total 6 MiB scratch

    const double Ad = pow(0.95, (double)CHUNK);      // 0.95^128
    const float  A    = (float)Ad;
    const float  A8   = (float)pow(Ad, 8.0);
    const float  A16  = (float)pow(Ad, 16.0);        // ~0 (true value underflows)
    const float  l2A  = (float)(log(Ad) / log(2.0));

    dim3 blk(256), grd(D_DIM / 256, NCHUNK);
    dim3 cblk(256), cgrd(D_DIM / 16 / 8);

    k_chunk_stats<<<grd, blk, 0, stream>>>(x, ws_sum, ws_sum2, ws_M);
    k_combine_wmma<<<cgrd, cblk, 0, stream>>>(ws_sum, ws_sum2, ws_M, ws_muin,
                                              0, A, A8, A16, l2A);
    k_var_agg<<<grd, blk, 0, stream>>>(x, ws_muin, ws_V);
    k_combine_wmma<<<cgrd, cblk, 0, stream>>>(ws_sum, ws_sum2, ws_V, ws_varin,
                                              1, A, A8, A16, l2A);
    k_final<<<grd, blk, 0, stream>>>(x, ws_muin, ws_varin, out_norm, out_info);
}